// Encoder_85229331021973
// MI455X (gfx1250) — compile-verified
//
#include <hip/hip_runtime.h>

typedef float v2f __attribute__((ext_vector_type(2)));
typedef float v8f __attribute__((ext_vector_type(8)));

// ---------------------------------------------------------------- utilities
__global__ __launch_bounds__(256) void k_zero4(float4* __restrict__ p, int n4) {
  int i = blockIdx.x * 256 + threadIdx.x;
  if (i < n4) p[i] = make_float4(0.f, 0.f, 0.f, 0.f);
}

__global__ __launch_bounds__(256) void k_degree(const int* __restrict__ dst,
                                                float* __restrict__ deg, int E) {
  int e = blockIdx.x * 256 + threadIdx.x;
  if (e < E) atomicAdd(&deg[dst[e]], 1.0f);
}

// dinv[i] = rsqrt(deg[i] + 1)   (+1 = self loop; guarantees >= 1)
__global__ __launch_bounds__(256) void k_rsqrt_deg(float* __restrict__ deg, int N) {
  int i = blockIdx.x * 256 + threadIdx.x;
  if (i < N) deg[i] = rsqrtf(deg[i] + 1.0f);
}

// ---------------------------------------------------------------- WMMA GEMM
// out[N x F] = A[N x K] @ W[K x F], f32 via V_WMMA_F32_16X16X4_F32.
// 256 threads = 8 waves; W staged in LDS; each wave computes a 16-row strip.
template <int K, int F>
__global__ __launch_bounds__(256) void k_gemm_wmma(const float* __restrict__ A,
                                                   const float* __restrict__ W,
                                                   float* __restrict__ out,
                                                   int n_rows) {
  __shared__ float wlds[K * F];
  const int tid = threadIdx.x;
  {
    const float4* __restrict__ Wv = reinterpret_cast<const float4*>(W);
    float4* Lv = reinterpret_cast<float4*>(wlds);
    constexpr int total4 = (K * F) / 4;  // multiple of 256 for our sizes
#pragma unroll
    for (int i = 0; i < total4 / 256; ++i) Lv[i * 256 + tid] = Wv[i * 256 + tid];
  }
  __syncthreads();

  const int wave = tid >> 5;
  const int lane = tid & 31;
  const int row_base = (blockIdx.x * 8 + wave) * 16;
  if (row_base >= n_rows) return;  // wave-uniform: EXEC stays all-ones

  // A fragment mapping (16x4 f32): lanes 0-15 -> K pair {0,1}, lanes 16-31 -> {2,3}
  const int m = lane & 15;
  const int kb = (lane >> 4) * 2;  // 0 or 2
  const float* __restrict__ arow = A + (size_t)(row_base + m) * K + kb;

  v8f acc[F / 16];
  {
    v8f zv = {};
#pragma unroll
    for (int nt = 0; nt < F / 16; ++nt) acc[nt] = zv;
  }

#pragma unroll 2
  for (int k = 0; k < K; k += 4) {
    v2f a;
    a[0] = arow[k];
    a[1] = arow[k + 1];
#pragma unroll
    for (int nt = 0; nt < F / 16; ++nt) {
      v2f b;
      b[0] = wlds[(k + kb) * F + nt * 16 + m];
      b[1] = wlds[(k + kb + 1) * F + nt * 16 + m];
      acc[nt] = __builtin_amdgcn_wmma_f32_16x16x4_f32(
          /*neg_a=*/false, a, /*neg_b=*/false, b,
          /*c_mod=*/(short)0, acc[nt], /*reuse_a=*/false, /*reuse_b=*/false);
    }
  }

  // C/D layout: VGPR r, lanes 0-15 -> M=r; lanes 16-31 -> M=8+r; N = lane&15
  const int mh = (lane >> 4) * 8;
#pragma unroll
  for (int nt = 0; nt < F / 16; ++nt) {
#pragma unroll
    for (int r = 0; r < 8; ++r) {
      out[(size_t)(row_base + mh + r) * F + nt * 16 + m] = acc[nt][r];
    }
  }
}

// ---------------------------------------------------------------- aggregation
// out[dst] += h[src] * dinv[src]*dinv[dst]; tasks e >= E are self loops.
// One thread per (edge, float4 chunk) -> fully coalesced over features.
template <int LOG2F4>
__global__ __launch_bounds__(256) void k_aggregate(const float* __restrict__ h,
                                                   const int* __restrict__ src,
                                                   const int* __restrict__ dst,
                                                   const float* __restrict__ dinv,
                                                   float* __restrict__ out,
                                                   int E, int N) {
  constexpr int F4 = 1 << LOG2F4;
  constexpr int F = F4 * 4;
  int t = blockIdx.x * 256 + threadIdx.x;
  int total = (E + N) << LOG2F4;
  if (t >= total) return;
  int e = t >> LOG2F4;
  int f4 = t & (F4 - 1);
  int s, d;
  float w;
  if (e < E) {
    s = src[e];
    d = dst[e];
    w = dinv[s] * dinv[d];
  } else {
    s = e - E;
    d = s;
    float di = dinv[s];
    w = di * di;
  }
  const float4 v = reinterpret_cast<const float4*>(h + (size_t)s * F)[f4];
  float* o = out + (size_t)d * F + (size_t)f4 * 4;
  atomicAdd(o + 0, v.x * w);
  atomicAdd(o + 1, v.y * w);
  atomicAdd(o + 2, v.z * w);
  atomicAdd(o + 3, v.w * w);
}

__global__ __launch_bounds__(256) void k_bias_relu(float* __restrict__ h,
                                                   const float* __restrict__ b,
                                                   int total, int fmask) {
  int i = blockIdx.x * 256 + threadIdx.x;
  if (i < total) {
    float v = h[i] + b[i & fmask];
    h[i] = v > 0.0f ? v : 0.0f;
  }
}

// ---------------------------------------------------------------- launcher
extern "C" void kernel_launch(void* const* d_in, const int* in_sizes, int n_in,
                              void* d_out, int out_size, void* d_ws, size_t ws_size,
                              hipStream_t stream) {
  (void)n_in; (void)out_size; (void)ws_size;

  const float* x  = (const float*)d_in[0];
  const float* W1 = (const float*)d_in[1];
  const float* b1 = (const float*)d_in[2];
  const float* W2 = (const float*)d_in[3];
  const float* b2 = (const float*)d_in[4];
  const int*   ei = (const int*)d_in[5];

  const int D_IN = 128, D_HID = 128, D_OUT = 64;
  const int N = in_sizes[0] / D_IN;
  const int E = in_sizes[5] / 2;
  const int* src = ei;        // edge_index[0, :]
  const int* dst = ei + E;    // edge_index[1, :]

  // workspace carve-up (all float4 aligned): dinv | bufA (N*128) | bufB (N*128)
  float* dinv = (float*)d_ws;
  float* bufA = dinv + (size_t)((N + 3) & ~3);
  float* bufB = bufA + (size_t)N * D_HID;
  float* out  = (float*)d_out;

  auto cdiv = [](long a, long b) { return (int)((a + b - 1) / b); };

  // ---- normalization coefficients
  k_zero4<<<cdiv(N / 4, 256), 256, 0, stream>>>((float4*)dinv, N / 4);
  k_degree<<<cdiv(E, 256), 256, 0, stream>>>(dst, dinv, E);
  k_rsqrt_deg<<<cdiv(N, 256), 256, 0, stream>>>(dinv, N);

  // ---- layer 1: H = relu( Agg(X @ W1) + b1 )
  k_gemm_wmma<128, 128><<<cdiv(N, 128), 256, 0, stream>>>(x, W1, bufA, N);
  k_zero4<<<cdiv((long)N * D_HID / 4, 256), 256, 0, stream>>>((float4*)bufB,
                                                              N * D_HID / 4);
  k_aggregate<5><<<cdiv((long)(E + N) * 32, 256), 256, 0, stream>>>(
      bufA, src, dst, dinv, bufB, E, N);
  k_bias_relu<<<cdiv((long)N * D_HID, 256), 256, 0, stream>>>(bufB, b1,
                                                              N * D_HID, D_HID - 1);

  // ---- layer 2: out = relu( Agg(H @ W2) + b2 )
  k_gemm_wmma<128, 64><<<cdiv(N, 128), 256, 0, stream>>>(bufB, W2, bufA, N);
  k_zero4<<<cdiv((long)N * D_OUT / 4, 256), 256, 0, stream>>>((float4*)out,
                                                              N * D_OUT / 4);
  k_aggregate<4><<<cdiv((long)(E + N) * 16, 256), 256, 0, stream>>>(
      bufA, src, dst, dinv, out, E, N);
  k_bias_relu<<<cdiv((long)N * D_OUT, 256), 256, 0, stream>>>(out, b2,
                                                              N * D_OUT, D_OUT - 1);
}